// GCNEncoder_32684701122703
// MI455X (gfx1250) — compile-verified
//
#include <hip/hip_runtime.h>

typedef float v2f __attribute__((ext_vector_type(2)));
typedef float v8f __attribute__((ext_vector_type(8)));

#define NNODES 10000
#define NEDGES 320000
#define DIN 512
#define DOUT 256

// ---------------------------------------------------------------------------
// h = X @ W0 + b0      X:[N,512]  W0:[512,256]  h:[N,256]
// Each wave computes a 16x32 tile (two 16x16 col tiles, shared A fragment,
// two independent accumulator chains). grid = (625,1), block = 256 (8 waves).
// WMMA f32 16x16x4 layouts (ISA 7.12.2):
//   A frag: lane16 = M, elem j -> K = k + 2*hsel + j
//   B frag: lane16 = N, elem j -> K = k + 2*hsel + j
//   C vgpr r -> row (r + 8*hsel), col lane16
// ---------------------------------------------------------------------------
__global__ __launch_bounds__(256) void lin_gemm_kernel(
    const float* __restrict__ X, const float* __restrict__ W,
    const float* __restrict__ bias, float* __restrict__ H)
{
    const int lane   = threadIdx.x & 31;
    const int wave   = threadIdx.x >> 5;
    const int lane16 = lane & 15;
    const int hsel   = lane >> 4;   // 0 or 1

    const int row0 = blockIdx.x * 16;   // 625 row tiles
    const int col0 = wave * 32;         // 8 waves cover all 256 cols

    const float* arow  = X + (size_t)(row0 + lane16) * DIN + 2 * hsel;
    const float* bcol0 = W + (size_t)(2 * hsel) * DOUT + (col0 + lane16);
    const float* bcol1 = bcol0 + 16;

    v8f acc0 = {};
    v8f acc1 = {};
    #pragma unroll 4
    for (int k = 0; k < DIN; k += 4) {
        v2f a = *(const v2f*)arow;               // one b64 load feeds 2 WMMAs
        v2f b0f, b1f;
        b0f.x = bcol0[0];
        b0f.y = bcol0[DOUT];
        b1f.x = bcol1[0];
        b1f.y = bcol1[DOUT];
        acc0 = __builtin_amdgcn_wmma_f32_16x16x4_f32(false, a, false, b0f,
                                                     (short)0, acc0, false, false);
        acc1 = __builtin_amdgcn_wmma_f32_16x16x4_f32(false, a, false, b1f,
                                                     (short)0, acc1, false, false);
        arow += 4;
        bcol0 += 4 * DOUT;
        bcol1 += 4 * DOUT;
    }

    const float bv0 = bias[col0 + lane16];
    const float bv1 = bias[col0 + 16 + lane16];
    float* out = H + (size_t)row0 * DOUT + col0 + lane16;
    #pragma unroll
    for (int r = 0; r < 8; ++r) {
        const size_t m = (size_t)(r + 8 * hsel) * DOUT;
        out[m]      = acc0[r] + bv0;
        out[m + 16] = acc1[r] + bv1;
    }
}

// ---------------------------------------------------------------------------
// out = [relu]( (agg*invdeg) @ Wl + bl + Hin @ Wr ) + Hin
// 16x32 tile per wave, two A streams (mean, h), 4 WMMAs per K step. K = 256.
// ---------------------------------------------------------------------------
__global__ __launch_bounds__(256) void sage_gemm_kernel(
    const float* __restrict__ AGG, const float* __restrict__ INVDEG,
    const float* __restrict__ Hin,
    const float* __restrict__ Wl, const float* __restrict__ bl,
    const float* __restrict__ Wr,
    float* __restrict__ OUT, int relu)
{
    const int lane   = threadIdx.x & 31;
    const int wave   = threadIdx.x >> 5;
    const int lane16 = lane & 15;
    const int hsel   = lane >> 4;

    const int row0 = blockIdx.x * 16;
    const int col0 = wave * 32;

    const int arow_idx = row0 + lane16;
    const float inv = INVDEG[arow_idx];

    const float* aggrow = AGG + (size_t)arow_idx * DOUT + 2 * hsel;
    const float* hrow   = Hin + (size_t)arow_idx * DOUT + 2 * hsel;
    const float* blc0   = Wl + (size_t)(2 * hsel) * DOUT + (col0 + lane16);
    const float* blc1   = blc0 + 16;
    const float* brc0   = Wr + (size_t)(2 * hsel) * DOUT + (col0 + lane16);
    const float* brc1   = brc0 + 16;

    v8f acc0 = {};
    v8f acc1 = {};
    #pragma unroll 2
    for (int k = 0; k < DOUT; k += 4) {
        v2f am = *(const v2f*)aggrow;
        am.x *= inv; am.y *= inv;                // mean = agg / max(deg,1)
        v2f ah = *(const v2f*)hrow;
        v2f bl0f, bl1f, br0f, br1f;
        bl0f.x = blc0[0];    bl0f.y = blc0[DOUT];
        bl1f.x = blc1[0];    bl1f.y = blc1[DOUT];
        br0f.x = brc0[0];    br0f.y = brc0[DOUT];
        br1f.x = brc1[0];    br1f.y = brc1[DOUT];
        acc0 = __builtin_amdgcn_wmma_f32_16x16x4_f32(false, am, false, bl0f,
                                                     (short)0, acc0, false, false);
        acc1 = __builtin_amdgcn_wmma_f32_16x16x4_f32(false, am, false, bl1f,
                                                     (short)0, acc1, false, false);
        acc0 = __builtin_amdgcn_wmma_f32_16x16x4_f32(false, ah, false, br0f,
                                                     (short)0, acc0, false, false);
        acc1 = __builtin_amdgcn_wmma_f32_16x16x4_f32(false, ah, false, br1f,
                                                     (short)0, acc1, false, false);
        aggrow += 4; hrow += 4;
        blc0 += 4 * DOUT; blc1 += 4 * DOUT;
        brc0 += 4 * DOUT; brc1 += 4 * DOUT;
    }

    const float bv0 = bl[col0 + lane16];
    const float bv1 = bl[col0 + 16 + lane16];
    const float* hres = Hin + (size_t)row0 * DOUT + col0 + lane16;
    float* out = OUT + (size_t)row0 * DOUT + col0 + lane16;
    #pragma unroll
    for (int r = 0; r < 8; ++r) {
        const size_t m = (size_t)(r + 8 * hsel) * DOUT;
        float v0 = acc0[r] + bv0;
        float v1 = acc1[r] + bv1;
        if (relu) { v0 = fmaxf(v0, 0.0f); v1 = fmaxf(v1, 0.0f); }
        out[m]      = v0 + hres[m];
        out[m + 16] = v1 + hres[m + 16];
    }
}

// ---------------------------------------------------------------------------
// agg[dst[e]] += H[src[e]]   one wave per edge, 8 f32 atomics per lane.
// ---------------------------------------------------------------------------
__global__ __launch_bounds__(256) void scatter_add_kernel(
    const float* __restrict__ H, const int* __restrict__ src,
    const int* __restrict__ dst, float* __restrict__ AGG, int E)
{
    const int e = blockIdx.x * 8 + (threadIdx.x >> 5);
    if (e >= E) return;
    const int lane = threadIdx.x & 31;
    const int s = src[e];
    const int d = dst[e];
    const float* hs = H + (size_t)s * DOUT;
    float* ad = AGG + (size_t)d * DOUT;
    #pragma unroll
    for (int j = 0; j < 8; ++j) {
        atomicAdd(&ad[lane + 32 * j], hs[lane + 32 * j]);
    }
}

__global__ void deg_kernel(const int* __restrict__ dst, float* __restrict__ deg, int E)
{
    const int e = blockIdx.x * blockDim.x + threadIdx.x;
    if (e < E) atomicAdd(&deg[dst[e]], 1.0f);
}

__global__ void invdeg_kernel(float* __restrict__ deg, int n)
{
    const int i = blockIdx.x * blockDim.x + threadIdx.x;
    if (i < n) deg[i] = 1.0f / fmaxf(deg[i], 1.0f);
}

__global__ void zero_kernel(float* __restrict__ p, int n)
{
    const int i = blockIdx.x * blockDim.x + threadIdx.x;
    if (i < n) p[i] = 0.0f;
}

// ---------------------------------------------------------------------------
extern "C" void kernel_launch(void* const* d_in, const int* in_sizes, int n_in,
                              void* d_out, int out_size, void* d_ws, size_t ws_size,
                              hipStream_t stream)
{
    const float* x   = (const float*)d_in[0];
    const int*   ei  = (const int*)d_in[1];
    const float* W0  = (const float*)d_in[2];
    const float* b0  = (const float*)d_in[3];
    const float* Wl1 = (const float*)d_in[4];
    const float* bl1 = (const float*)d_in[5];
    const float* Wr1 = (const float*)d_in[6];
    const float* Wl2 = (const float*)d_in[7];
    const float* bl2 = (const float*)d_in[8];
    const float* Wr2 = (const float*)d_in[9];

    const int* src = ei;             // edge_index[0, :]
    const int* dst = ei + NEDGES;    // edge_index[1, :]

    float* ws  = (float*)d_ws;
    float* h   = ws;                               // N * 256
    float* agg = h + (size_t)NNODES * DOUT;        // N * 256
    float* deg = agg + (size_t)NNODES * DOUT;      // N

    float* out1 = (float*)d_out;
    float* out2 = out1 + (size_t)NNODES * DOUT;

    const int feat_elems = NNODES * DOUT;
    dim3 gemm_grid(NNODES / 16, 1);   // 625 blocks, 8 waves, 16x32 tile/wave

    // init scratch (harness poisons it; deterministic zeroing every call)
    zero_kernel<<<(feat_elems + 255) / 256, 256, 0, stream>>>(agg, feat_elems);
    zero_kernel<<<(NNODES + 255) / 256, 256, 0, stream>>>(deg, NNODES);

    // h = x @ W0 + b0
    lin_gemm_kernel<<<gemm_grid, 256, 0, stream>>>(x, W0, b0, h);

    // degrees -> invdeg (shared by both layers)
    deg_kernel<<<(NEDGES + 255) / 256, 256, 0, stream>>>(dst, deg, NEDGES);
    invdeg_kernel<<<(NNODES + 255) / 256, 256, 0, stream>>>(deg, NNODES);

    // layer 1: aggregate h then fused dual-GEMM (+bias, relu, residual)
    scatter_add_kernel<<<NEDGES / 8, 256, 0, stream>>>(h, src, dst, agg, NEDGES);
    sage_gemm_kernel<<<gemm_grid, 256, 0, stream>>>(agg, deg, h, Wl1, bl1, Wr1, out1, 1);

    // layer 2: re-zero, aggregate out1, fused dual-GEMM (no relu)
    zero_kernel<<<(feat_elems + 255) / 256, 256, 0, stream>>>(agg, feat_elems);
    scatter_add_kernel<<<NEDGES / 8, 256, 0, stream>>>(out1, src, dst, agg, NEDGES);
    sage_gemm_kernel<<<gemm_grid, 256, 0, stream>>>(agg, deg, out1, Wl2, bl2, Wr2, out2, 0);
}